// SplineConv_84189948936233
// MI455X (gfx1250) — compile-verified
//
#include <hip/hip_runtime.h>

// ---- problem constants (match reference) ----
#define NK      5          // kernel size per dim
#define K_TOT   25         // 5^2 flattened kernel slots
#define M_IN    32
#define M_OUT   32
#define KH      (K_TOT * M_IN)   // 800 : flattened GEMM K dimension

typedef __attribute__((ext_vector_type(2))) float v2f;
typedef __attribute__((ext_vector_type(8))) float v8f;

// ---------------------------------------------------------------------------
// Kernel A: zero the workspace (h + deg) with float4 stores, grid-stride.
// ---------------------------------------------------------------------------
__global__ void spline_zero_kernel(float4* __restrict__ p, size_t n4,
                                   float* __restrict__ tail, size_t ntail) {
    size_t i = (size_t)blockIdx.x * blockDim.x + threadIdx.x;
    size_t stride = (size_t)gridDim.x * blockDim.x;
    float4 z; z.x = 0.f; z.y = 0.f; z.z = 0.f; z.w = 0.f;
    for (; i < n4; i += stride) p[i] = z;
    size_t t = (size_t)blockIdx.x * blockDim.x + threadIdx.x;
    if (t < ntail) tail[t] = 0.f;
}

// ---------------------------------------------------------------------------
// Kernel A2: pack weight/root_weight into f32 WMMA B-fragment order.
// For K-step ks (4 K-values), tile (out-cols 0-15 / 16-31), lane, vgpr v:
//   wpk[((ks*2 + tile)*32 + lane)*2 + v] = w[(4*ks + 2*hi + v)*32 + tile*16+n]
// where hi = lane>>4, n = lane&15.  The hot loop then fetches each B
// fragment as one coalesced global_load_b64.
// ---------------------------------------------------------------------------
__global__ void spline_pack_kernel(const float* __restrict__ w,
                                   const float* __restrict__ rw,
                                   float* __restrict__ wpk,
                                   float* __restrict__ rpk) {
    const int nW = (KH / 4) * 128;    // 25600
    const int nR = (M_IN / 4) * 128;  // 1024
    int t = blockIdx.x * blockDim.x + threadIdx.x;
    if (t < nW) {
        int ks = t >> 7, r = t & 127;
        int tile = r >> 6, lane = (r >> 1) & 31, v = r & 1;
        int hi = lane >> 4, n = lane & 15;
        wpk[t] = w[(4 * ks + 2 * hi + v) * M_OUT + tile * 16 + n];
    } else if (t < nW + nR) {
        int u = t - nW;
        int ks = u >> 7, r = u & 127;
        int tile = r >> 6, lane = (r >> 1) & 31, v = r & 1;
        int hi = lane >> 4, n = lane & 15;
        rpk[u] = rw[(4 * ks + 2 * hi + v) * M_OUT + tile * 16 + n];
    }
}

// ---------------------------------------------------------------------------
// Kernel B: one wave32 per edge, lane = input channel.
// Degree-1 tensor-product spline basis (4 corners), gather source feature,
// scatter basis-weighted features into h with coalesced global_atomic_add_f32.
// ---------------------------------------------------------------------------
__global__ void spline_scatter_kernel(const float* __restrict__ x,
                                      const long long* __restrict__ ei,
                                      const float* __restrict__ pseudo,
                                      float* __restrict__ h,
                                      float* __restrict__ deg,
                                      int nE) {
    int wave = (int)(((size_t)blockIdx.x * blockDim.x + threadIdx.x) >> 5);
    int lane = threadIdx.x & 31;
    if (wave >= nE) return;
    int e = wave;

    int row = (int)ei[e];                 // target
    int col = (int)ei[(size_t)nE + e];    // source

    float p0 = pseudo[2 * e]     * (float)(NK - 1);
    float p1 = pseudo[2 * e + 1] * (float)(NK - 1);
    float l0 = floorf(p0), l1 = floorf(p1);
    float f0 = p0 - l0,    f1 = p1 - l1;
    float g0 = 1.0f - f0,  g1 = 1.0f - f1;
    int i0 = (int)l0, i1 = (int)l1;
    int i00 = min(max(i0,     0), NK - 1);
    int i01 = min(max(i0 + 1, 0), NK - 1);
    int i10 = min(max(i1,     0), NK - 1);
    int i11 = min(max(i1 + 1, 0), NK - 1);

    float xv = x[(size_t)col * M_IN + lane];
    float* hrow = h + (size_t)row * KH;

    atomicAdd(hrow + ((size_t)(i00 + NK * i10)) * M_IN + lane, g0 * g1 * xv);
    atomicAdd(hrow + ((size_t)(i01 + NK * i10)) * M_IN + lane, f0 * g1 * xv);
    atomicAdd(hrow + ((size_t)(i00 + NK * i11)) * M_IN + lane, g0 * f1 * xv);
    atomicAdd(hrow + ((size_t)(i01 + NK * i11)) * M_IN + lane, f0 * f1 * xv);

    if (lane == 0) atomicAdd(deg + row, 1.0f);
}

// ---------------------------------------------------------------------------
// Kernel C: fp32 WMMA GEMM with 2x M register blocking.
//   out = (h[N,800] @ w[800,32]) / deg + x @ rw + bias
// One wave per 32-node pair-tile: 2 A fragments x 2 B fragments -> 4 v8f
// accumulators.  Hot loop per K-step: 2x global_load_b64 (A, streams h)
// + 2x global_load_b64 (packed B, cache-resident) feeding 4 WMMAs.
// ---------------------------------------------------------------------------
__global__ void spline_gemm_kernel(const float* __restrict__ h,
                                   const float* __restrict__ deg,
                                   const float* __restrict__ x,
                                   const float* __restrict__ wpk,  // packed B
                                   const float* __restrict__ rpk,  // packed root B
                                   const float* __restrict__ bias,
                                   float* __restrict__ out,
                                   int n) {
    int wave = (int)(((size_t)blockIdx.x * blockDim.x + threadIdx.x) >> 5);
    int lane = threadIdx.x & 31;
    int ntiles = (n + 15) >> 4;
    int npair = (ntiles + 1) >> 1;
    if (wave >= npair) return;

    int n0 = wave << 5;          // first node of the 32-node pair-tile
    int hi = lane >> 4;
    int m  = lane & 15;

    int rowA = n0 + m;      if (rowA >= n) rowA = n - 1;   // clamp: EXEC full
    int rowB = n0 + 16 + m; if (rowB >= n) rowB = n - 1;

    const float* aA = h + (size_t)rowA * KH;
    const float* aB = h + (size_t)rowB * KH;

    v8f c00 = {}, c01 = {};   // nodes n0..n0+15   x cols 0-15 / 16-31
    v8f c10 = {}, c11 = {};   // nodes n0+16..+31  x cols 0-15 / 16-31

    for (int k0 = 0; k0 < KH; k0 += 4) {
        v2f a0 = *(const v2f*)(aA + k0 + 2 * hi);
        v2f a1 = *(const v2f*)(aB + k0 + 2 * hi);
        const float* bp = wpk + ((size_t)k0 << 5);    // (k0/4)*128
        v2f b0 = *(const v2f*)(bp + 2 * lane);
        v2f b1 = *(const v2f*)(bp + 64 + 2 * lane);
        c00 = __builtin_amdgcn_wmma_f32_16x16x4_f32(false, a0, false, b0,
                                                    (short)0, c00, false, false);
        c01 = __builtin_amdgcn_wmma_f32_16x16x4_f32(false, a0, false, b1,
                                                    (short)0, c01, false, false);
        c10 = __builtin_amdgcn_wmma_f32_16x16x4_f32(false, a1, false, b0,
                                                    (short)0, c10, false, false);
        c11 = __builtin_amdgcn_wmma_f32_16x16x4_f32(false, a1, false, b1,
                                                    (short)0, c11, false, false);
    }

    // mean aggregation: scale each output row by 1/max(deg,1)
    int rbase0 = n0 + 8 * hi;
    int rbase1 = n0 + 16 + 8 * hi;
#pragma unroll
    for (int r = 0; r < 8; ++r) {
        int rr0 = rbase0 + r;
        float d0 = (rr0 < n) ? deg[rr0] : 1.0f;
        float s0 = 1.0f / fmaxf(d0, 1.0f);
        c00[r] *= s0; c01[r] *= s0;
        int rr1 = rbase1 + r;
        float d1 = (rr1 < n) ? deg[rr1] : 1.0f;
        float s1 = 1.0f / fmaxf(d1, 1.0f);
        c10[r] *= s1; c11[r] *= s1;
    }

    // root term: 8 more fp32 WMMA K-steps folding x @ root_weight
    const float* xA = x + (size_t)rowA * M_IN;
    const float* xB = x + (size_t)rowB * M_IN;
    for (int k0 = 0; k0 < M_IN; k0 += 4) {
        v2f a0 = *(const v2f*)(xA + k0 + 2 * hi);
        v2f a1 = *(const v2f*)(xB + k0 + 2 * hi);
        const float* bp = rpk + ((size_t)k0 << 5);
        v2f b0 = *(const v2f*)(bp + 2 * lane);
        v2f b1 = *(const v2f*)(bp + 64 + 2 * lane);
        c00 = __builtin_amdgcn_wmma_f32_16x16x4_f32(false, a0, false, b0,
                                                    (short)0, c00, false, false);
        c01 = __builtin_amdgcn_wmma_f32_16x16x4_f32(false, a0, false, b1,
                                                    (short)0, c01, false, false);
        c10 = __builtin_amdgcn_wmma_f32_16x16x4_f32(false, a1, false, b0,
                                                    (short)0, c10, false, false);
        c11 = __builtin_amdgcn_wmma_f32_16x16x4_f32(false, a1, false, b1,
                                                    (short)0, c11, false, false);
    }

    float bb0 = bias[m];
    float bb1 = bias[16 + m];

#pragma unroll
    for (int r = 0; r < 8; ++r) {
        int rr0 = rbase0 + r;
        if (rr0 < n) {
            out[(size_t)rr0 * M_OUT + m]      = c00[r] + bb0;
            out[(size_t)rr0 * M_OUT + 16 + m] = c01[r] + bb1;
        }
        int rr1 = rbase1 + r;
        if (rr1 < n) {
            out[(size_t)rr1 * M_OUT + m]      = c10[r] + bb0;
            out[(size_t)rr1 * M_OUT + 16 + m] = c11[r] + bb1;
        }
    }
}

// ---------------------------------------------------------------------------
// Host-side launch. Inputs (setup_inputs order):
//   0: x           [N, 32]        float32
//   1: edge_index  [2, E]         int64
//   2: pseudo      [E, 2]         float32
//   3: weight      [25, 32, 32]   float32
//   4: root_weight [32, 32]       float32
//   5: bias        [32]           float32
// Output: out [N, 32] float32
// Workspace: h [N*800] f32 (320 MB) + deg [N] f32 + wpk [25600] + rpk [1024]
// ---------------------------------------------------------------------------
extern "C" void kernel_launch(void* const* d_in, const int* in_sizes, int n_in,
                              void* d_out, int out_size, void* d_ws, size_t ws_size,
                              hipStream_t stream) {
    const float*     x      = (const float*)d_in[0];
    const long long* ei     = (const long long*)d_in[1];
    const float*     pseudo = (const float*)d_in[2];
    const float*     w      = (const float*)d_in[3];
    const float*     rw     = (const float*)d_in[4];
    const float*     bias   = (const float*)d_in[5];
    float*           out    = (float*)d_out;

    int N = in_sizes[0] / M_IN;
    int E = in_sizes[1] / 2;

    float* h   = (float*)d_ws;
    float* deg = h + (size_t)N * KH;
    float* wpk = deg + (size_t)N;
    float* rpk = wpk + (KH / 4) * 128;

    // Phase A: zero h + deg
    size_t total = (size_t)N * KH + (size_t)N;
    size_t n4    = total >> 2;
    size_t ntail = total & 3;
    float* tail  = (float*)d_ws + (n4 << 2);
    {
        int threads = 256;
        size_t want = (n4 + threads - 1) / threads;
        int blocks = (int)((want < 65536) ? (want ? want : 1) : 65536);
        spline_zero_kernel<<<blocks, threads, 0, stream>>>((float4*)d_ws, n4,
                                                           tail, ntail);
    }

    // Phase A2: pack B fragments (26624 elements)
    {
        int nElem = (KH / 4) * 128 + (M_IN / 4) * 128;
        int threads = 256;
        int blocks = (nElem + threads - 1) / threads;
        spline_pack_kernel<<<blocks, threads, 0, stream>>>(w, rw, wpk, rpk);
    }

    // Phase B: edge scatter (one wave32 per edge)
    {
        int threads = 256;                       // 8 edges per block
        int blocks = (E + 7) / 8;
        spline_scatter_kernel<<<blocks, threads, 0, stream>>>(x, ei, pseudo,
                                                              h, deg, E);
    }

    // Phase C: WMMA GEMM + mean + root + bias (one wave per 32-node tile)
    {
        int ntiles = (N + 15) / 16;
        int npair  = (ntiles + 1) / 2;
        int threads = 256;                       // 8 pair-tiles per block
        int blocks = (npair + 7) / 8;
        spline_gemm_kernel<<<blocks, threads, 0, stream>>>(h, deg, x, wpk, rpk,
                                                           bias, out, N);
    }
}